// ColbertPairwiseNegativeCELoss_27350351741582
// MI455X (gfx1250) — compile-verified
//
#include <hip/hip_runtime.h>
#include <hip/hip_bf16.h>
#include <math.h>

// ColBERT pairwise-negative CE loss for MI455X (gfx1250, wave32).
// Q:(64,32,128) fp32, Dpos:(64,256,128) fp32, Dneg:(64,256,128) fp32.
// Core op mapped onto V_WMMA_F32_16X16X4_F32.
//
// Per (b,c) pair workgroup (8 waves): wave w owns s-tiles {w, w+8}; for each
// s-tile it loads the doc A-operand ONCE per K-step and issues two WMMAs
// (n-tile 0 and 1), halving global traffic vs. the naive 4-tile loop and
// providing two independent accumulation chains for the scheduler.

typedef float v2f __attribute__((ext_vector_type(2)));
typedef float v8f __attribute__((ext_vector_type(8)));

#define BATCH 64
#define NQ    32     // query tokens per batch
#define SD    256    // doc tokens per batch
#define DIM   128    // embedding dim

__global__ __launch_bounds__(256) void maxsim_scores_kernel(
    const float* __restrict__ q,      // (B, NQ, DIM)
    const float* __restrict__ doc,    // (B, SD, DIM)
    float* __restrict__ out,          // diag ? (B) : (B,B)
    int diag_only)
{
    __shared__ float qlds[NQ * DIM];        // 16 KB: Q[b] tile
    __shared__ float partial[8][NQ];        // per-wave per-n running max

    const int b = diag_only ? blockIdx.x : blockIdx.y;
    const int c = blockIdx.x;

    const int tid  = threadIdx.x;
    const int lane = tid & 31;
    const int wv   = tid >> 5;
    const int l15  = lane & 15;
    const int hi   = lane >> 4;          // lane half: K sub-pair / M-high rows

    // ---- Stage Q[b] (32x128 fp32 = 16KB) into LDS with b128 loads ----
    {
        const float4* qsrc = (const float4*)(q + (size_t)b * (NQ * DIM));
        float4* qdst = (float4*)qlds;
        #pragma unroll
        for (int i = tid; i < (NQ * DIM / 4); i += 256) qdst[i] = qsrc[i];
    }
    __syncthreads();

    const float* docbase = doc + (size_t)c * (SD * DIM);

    // B operands (Q^T): col n = l15 of each n-tile, K indexing k0 + 2*hi + {0,1}
    const float* qrow0 = qlds + (l15) * DIM + 2 * hi;          // n-tile 0
    const float* qrow1 = qlds + (16 + l15) * DIM + 2 * hi;     // n-tile 1

    // rm[nt] = running max over this wave's s-tiles, for query column n = l15
    float rm[2] = { -3.0e38f, -3.0e38f };

    #pragma unroll
    for (int sti = 0; sti < 2; ++sti) {
        const int st = wv + sti * 8;     // s-tile (0..15)

        // A operand (doc): row m = l15 of tile, cols k0 + 2*hi + {0,1}
        const float* drow = docbase + (size_t)(st * 16 + l15) * DIM + 2 * hi;

        v8f acc0 = {};                   // D tile for n-tile 0
        v8f acc1 = {};                   // D tile for n-tile 1
        #pragma unroll 8
        for (int k0 = 0; k0 < DIM; k0 += 4) {
            v2f a  = *(const v2f*)(drow  + k0);  // global_load_b64 (shared by both WMMAs)
            v2f b0 = *(const v2f*)(qrow0 + k0);  // ds_load_b64
            v2f b1 = *(const v2f*)(qrow1 + k0);  // ds_load_b64
            acc0 = __builtin_amdgcn_wmma_f32_16x16x4_f32(
                false, a, false, b0, (short)0, acc0, false, false);
            acc1 = __builtin_amdgcn_wmma_f32_16x16x4_f32(
                false, a, false, b1, (short)0, acc1, false, false);
        }

        // D layout: lane holds column n = l15; rows s = vgpr + 8*hi.
        float m0 = acc0[0];
        m0 = fmaxf(m0, acc0[1]); m0 = fmaxf(m0, acc0[2]); m0 = fmaxf(m0, acc0[3]);
        m0 = fmaxf(m0, acc0[4]); m0 = fmaxf(m0, acc0[5]); m0 = fmaxf(m0, acc0[6]);
        m0 = fmaxf(m0, acc0[7]);
        rm[0] = fmaxf(rm[0], m0);

        float m1 = acc1[0];
        m1 = fmaxf(m1, acc1[1]); m1 = fmaxf(m1, acc1[2]); m1 = fmaxf(m1, acc1[3]);
        m1 = fmaxf(m1, acc1[4]); m1 = fmaxf(m1, acc1[5]); m1 = fmaxf(m1, acc1[6]);
        m1 = fmaxf(m1, acc1[7]);
        rm[1] = fmaxf(rm[1], m1);
    }

    // Merge the two lane halves (rows s+0..7 vs s+8..15 of each tile)
    rm[0] = fmaxf(rm[0], __shfl_xor(rm[0], 16, 32));
    rm[1] = fmaxf(rm[1], __shfl_xor(rm[1], 16, 32));

    if (lane < 16) {
        partial[wv][l15]      = rm[0];
        partial[wv][16 + l15] = rm[1];
    }
    __syncthreads();

    // Final: max over 8 waves per n, then sum over the 32 n values.
    if (tid < 32) {
        float m = partial[0][tid];
        #pragma unroll
        for (int w = 1; w < 8; ++w) m = fmaxf(m, partial[w][tid]);
        // sum across all 32 lanes
        #pragma unroll
        for (int off = 16; off > 0; off >>= 1) m += __shfl_xor(m, off, 32);
        if (tid == 0) {
            if (diag_only) out[b] = m;
            else           out[(size_t)b * BATCH + c] = m;
        }
    }
}

__device__ __forceinline__ float softplus_f(float x) {
    // numerically stable: max(x,0) + log1p(exp(-|x|))
    return fmaxf(x, 0.0f) + log1pf(expf(-fabsf(x)));
}

__global__ __launch_bounds__(64) void finalize_kernel(
    const float* __restrict__ scores,     // (B,B)
    const float* __restrict__ neg_scores, // (B)
    float* __restrict__ out)              // scalar
{
    __shared__ float red[BATCH];
    const int b = threadIdx.x;

    const float pos = scores[(size_t)b * BATCH + b];

    // explicit-negative term
    const float l1 = softplus_f(neg_scores[b] - pos);

    // in-batch term: row max with diagonal suppressed by 1e6
    float mx = -3.0e38f;
    #pragma unroll 8
    for (int cc = 0; cc < BATCH; ++cc) {
        float v = scores[(size_t)b * BATCH + cc] - (cc == b ? 1000000.0f : 0.0f);
        mx = fmaxf(mx, v);
    }
    const float l2 = softplus_f(mx - pos);

    red[b] = l1 + l2;
    __syncthreads();
    #pragma unroll
    for (int s = 32; s > 0; s >>= 1) {
        if (b < s) red[b] += red[b + s];
        __syncthreads();
    }
    if (b == 0) out[0] = red[0] * (1.0f / 128.0f);  // (mean+mean)/2 = sum/(2*64)
}

extern "C" void kernel_launch(void* const* d_in, const int* in_sizes, int n_in,
                              void* d_out, int out_size, void* d_ws, size_t ws_size,
                              hipStream_t stream) {
    const float* q  = (const float*)d_in[0];  // (64,32,128)
    const float* dp = (const float*)d_in[1];  // (64,256,128)
    const float* dn = (const float*)d_in[2];  // (64,256,128)

    float* scores = (float*)d_ws;             // 64*64 floats
    float* negs   = scores + BATCH * BATCH;   // 64 floats

    // all-pairs scores[b,c]
    maxsim_scores_kernel<<<dim3(BATCH, BATCH), 256, 0, stream>>>(q, dp, scores, 0);
    // per-batch negative scores (diagonal pairs only)
    maxsim_scores_kernel<<<dim3(BATCH, 1), 256, 0, stream>>>(q, dn, negs, 1);
    // losses + reduction
    finalize_kernel<<<1, BATCH, 0, stream>>>(scores, negs, (float*)d_out);
}